// SpatialTransformerLayer_49649821941825
// MI455X (gfx1250) — compile-verified
//
#include <hip/hip_runtime.h>

typedef __attribute__((ext_vector_type(2))) float v2f;
typedef __attribute__((ext_vector_type(8))) float v8f;

#define Bsz 16
#define Hh  512
#define Ww  1024
#define Nn  (Hh * Ww)   // 524288 pixels per batch

struct __attribute__((aligned(4))) F3 { float x, y, z; };

// ---------------------------------------------------------------------------
// Zero the depth-map region of d_out (harness poisons it with 0xAA).
// ---------------------------------------------------------------------------
__global__ void zero_kernel(float4* __restrict__ p, int n4) {
    int i = blockIdx.x * blockDim.x + threadIdx.x;
    int stride = gridDim.x * blockDim.x;
    float4 z = make_float4(0.f, 0.f, 0.f, 0.f);
    for (; i < n4; i += stride) p[i] = z;
}

// 3x3 * 3x3 (row-major)
__device__ inline void mat33_mul(const float* X, const float* Y, float* Z) {
#pragma unroll
    for (int i = 0; i < 3; ++i)
#pragma unroll
        for (int j = 0; j < 3; ++j)
            Z[i * 3 + j] = X[i * 3 + 0] * Y[0 * 3 + j]
                         + X[i * 3 + 1] * Y[1 * 3 + j]
                         + X[i * 3 + 2] * Y[2 * 3 + j];
}

// ---------------------------------------------------------------------------
// Main kernel: one batch per blockIdx.y. Each wave32 processes 32 pixels per
// iteration: two 16-pixel B-matrices through V_WMMA_F32_16X16X4_F32
//   du = A * [x,y,1,0]^T   (A = R * K^-1)    -> pts  = du*z + t
//   dv = M * [x,y,1,0]^T   (M = K * A)       -> proj = dv*z + Kt
// Tile-1 results are shuffled to lanes 16..31 via ds_bpermute so the whole
// wave runs the epilogue (full 128B coalesced z loads / cloud stores).
// ---------------------------------------------------------------------------
__global__ __launch_bounds__(256)
void st_kernel(const float* __restrict__ quat,   // (B,4)
               const float* __restrict__ radar,  // (B,1,H,W)
               const float* __restrict__ kmat,   // (B,3,3)
               const float* __restrict__ trans,  // (B,3)
               float* __restrict__ depth_out,    // (B,H,W)
               float* __restrict__ cloud_out) {  // (B,N,3)
    __shared__ float sA[9], sM[9], sT[3], sC[3];

    const int b = blockIdx.y;

    if (threadIdx.x == 0) {
        // --- normalize quaternion, build R ---
        float qw = quat[b * 4 + 0], qx = quat[b * 4 + 1];
        float qy = quat[b * 4 + 2], qz = quat[b * 4 + 3];
        float qn = rsqrtf(qw * qw + qx * qx + qy * qy + qz * qz);
        qw *= qn; qx *= qn; qy *= qn; qz *= qn;
        float R[9];
        R[0] = 1.f - 2.f * (qy * qy + qz * qz);
        R[1] = 2.f * (qx * qy - qw * qz);
        R[2] = 2.f * (qx * qz + qw * qy);
        R[3] = 2.f * (qx * qy + qw * qz);
        R[4] = 1.f - 2.f * (qx * qx + qz * qz);
        R[5] = 2.f * (qy * qz - qw * qx);
        R[6] = 2.f * (qx * qz - qw * qy);
        R[7] = 2.f * (qy * qz + qw * qx);
        R[8] = 1.f - 2.f * (qx * qx + qy * qy);

        // --- general 3x3 inverse of K ---
        const float* K = kmat + b * 9;
        float a = K[0], bb = K[1], c = K[2];
        float d = K[3], e  = K[4], f = K[5];
        float g = K[6], h  = K[7], i = K[8];
        float det = a * (e * i - f * h) - bb * (d * i - f * g) + c * (d * h - e * g);
        float id  = 1.f / det;
        float Kinv[9];
        Kinv[0] =  (e * i - f * h) * id;
        Kinv[1] = -(bb * i - c * h) * id;
        Kinv[2] =  (bb * f - c * e) * id;
        Kinv[3] = -(d * i - f * g) * id;
        Kinv[4] =  (a * i - c * g) * id;
        Kinv[5] = -(a * f - c * d) * id;
        Kinv[6] =  (d * h - e * g) * id;
        Kinv[7] = -(a * h - bb * g) * id;
        Kinv[8] =  (a * e - bb * d) * id;

        float A[9], M[9];
        mat33_mul(R, Kinv, A);   // A = R * K^-1
        mat33_mul(K, A, M);      // M = K * A
#pragma unroll
        for (int k = 0; k < 9; ++k) { sA[k] = A[k]; sM[k] = M[k]; }
        float t0 = trans[b * 3 + 0], t1 = trans[b * 3 + 1], t2 = trans[b * 3 + 2];
        sT[0] = t0; sT[1] = t1; sT[2] = t2;
        sC[0] = K[0] * t0 + K[1] * t1 + K[2] * t2;   // c = K * t
        sC[1] = K[3] * t0 + K[4] * t1 + K[5] * t2;
        sC[2] = K[6] * t0 + K[7] * t1 + K[8] * t2;
    }
    __syncthreads();

    const int lane = threadIdx.x & 31;
    const int wave = threadIdx.x >> 5;
    const int m    = lane & 15;     // A-matrix row for this lane
    const int kh   = lane >> 4;     // K half: 0 -> K{0,1}, 1 -> K{2,3}
    const bool lo  = lane < 16;
    const int  pidx = (lane & 15) * 4;  // ds_bpermute byte index (loop-invariant)

    // Loop-invariant A-matrix operands (16x4, rows 0..2 hold the 3x3, rest 0).
    v2f Aa, Ma;
    Aa.x = (m < 3) ? sA[m * 3 + (kh ? 2 : 0)] : 0.f;   // K=0 or K=2
    Aa.y = (m < 3 && kh == 0) ? sA[m * 3 + 1] : 0.f;   // K=1 (K=3 is zero)
    Ma.x = (m < 3) ? sM[m * 3 + (kh ? 2 : 0)] : 0.f;
    Ma.y = (m < 3 && kh == 0) ? sM[m * 3 + 1] : 0.f;
    const float t0 = sT[0], t1 = sT[1], t2 = sT[2];
    const float c0 = sC[0], c1 = sC[1], c2 = sC[2];

    const float* __restrict__ z_b = radar + (size_t)b * Nn;
    float* __restrict__ depth_b   = depth_out + (size_t)b * Nn;
    float* __restrict__ cloud_b   = cloud_out + (size_t)b * (size_t)Nn * 3;

    const int gw = blockIdx.x * (blockDim.x >> 5) + wave;
    const int nw = gridDim.x * (blockDim.x >> 5);
    const int stride = nw * 32;

    for (int base = gw * 32; base < Nn; base += stride) {
        const int n0 = base + lane;          // this lane's pixel (epilogue)

        // Prefetch next tile's z line (global_prefetch_b8); clamp, no branch.
        {
            int np = n0 + stride; np = np < Nn ? np : Nn - 1;
            __builtin_prefetch(z_b + np, 0, 0);
        }

        // B-matrices (4x16): column = [x, y, 1, 0]. Lanes 0..15 carry K{0,1}
        // (x,y), lanes 16..31 carry K{2,3} (1,0).
        const int na = base + m;          // tile-0 pixel for lanes 0..15
        const int nb = base + 16 + m;     // tile-1 pixel
        v2f B0, B1;
        B0.x = lo ? (float)(na & (Ww - 1)) : 1.f;
        B0.y = lo ? (float)(na >> 10)      : 0.f;
        B1.x = lo ? (float)(nb & (Ww - 1)) : 1.f;
        B1.y = lo ? (float)(nb >> 10)      : 0.f;

        v8f cz = {};
        v8f du0 = __builtin_amdgcn_wmma_f32_16x16x4_f32(
            false, Aa, false, B0, (short)0, cz, false, false);
        v8f dv0 = __builtin_amdgcn_wmma_f32_16x16x4_f32(
            false, Ma, false, B0, (short)0, cz, false, false);
        v8f du1 = __builtin_amdgcn_wmma_f32_16x16x4_f32(
            false, Aa, false, B1, (short)0, cz, false, false);
        v8f dv1 = __builtin_amdgcn_wmma_f32_16x16x4_f32(
            false, Ma, false, B1, (short)0, cz, false, false);

        // Merge: lanes 0..15 keep tile-0 (col = lane); lanes 16..31 fetch
        // tile-1 col = lane-16 from the lower half via ds_bpermute.
        float u0, u1, u2, v0, v1, v2;
        {
            float s;
            s = __int_as_float(__builtin_amdgcn_ds_bpermute(pidx, __float_as_int(du1[0])));
            u0 = lo ? du0[0] : s;
            s = __int_as_float(__builtin_amdgcn_ds_bpermute(pidx, __float_as_int(du1[1])));
            u1 = lo ? du0[1] : s;
            s = __int_as_float(__builtin_amdgcn_ds_bpermute(pidx, __float_as_int(du1[2])));
            u2 = lo ? du0[2] : s;
            s = __int_as_float(__builtin_amdgcn_ds_bpermute(pidx, __float_as_int(dv1[0])));
            v0 = lo ? dv0[0] : s;
            s = __int_as_float(__builtin_amdgcn_ds_bpermute(pidx, __float_as_int(dv1[1])));
            v1 = lo ? dv0[1] : s;
            s = __int_as_float(__builtin_amdgcn_ds_bpermute(pidx, __float_as_int(dv1[2])));
            v2 = lo ? dv0[2] : s;
        }

        // ---- full-wave epilogue: one pixel per lane ----
        const float z = z_b[n0];                 // 128B coalesced
        const float ptx = fmaf(u0, z, t0);       // pts  = A*p*z + t
        const float pty = fmaf(u1, z, t1);
        const float ptz = fmaf(u2, z, t2);
        const float prx = fmaf(v0, z, c0);       // proj = M*p*z + K*t
        const float pry = fmaf(v1, z, c1);
        const float dep = fmaf(v2, z, c2);

        const float inv = __builtin_amdgcn_rcpf(dep);   // v_rcp_f32
        const int px = __float2int_rn(prx * inv);       // round-half-even
        const int py = __float2int_rn(pry * inv);
        const bool valid = (z > 0.f) & (px >= 0) & (px < Ww) &
                           (py >= 0) & (py < Hh);
        if (valid) depth_b[py * Ww + px] = dep;

        F3 out;
        out.x = valid ? ptx : 0.f;
        out.y = valid ? pty : 0.f;
        out.z = valid ? ptz : 0.f;
        *(F3*)(cloud_b + (size_t)n0 * 3) = out;   // single 12B store
    }
}

extern "C" void kernel_launch(void* const* d_in, const int* in_sizes, int n_in,
                              void* d_out, int out_size, void* d_ws, size_t ws_size,
                              hipStream_t stream) {
    (void)in_sizes; (void)n_in; (void)out_size; (void)d_ws; (void)ws_size;

    const float* quat  = (const float*)d_in[0];   // (B,4)
    const float* radar = (const float*)d_in[1];   // (B,1,H,W)
    const float* kmat  = (const float*)d_in[2];   // (B,3,3)
    const float* trans = (const float*)d_in[3];   // (B,3)

    float* out       = (float*)d_out;
    float* depth_out = out;                        // B*N floats
    float* cloud_out = out + (size_t)Bsz * Nn;     // B*N*3 floats

    // 1) zero the scatter target (depth maps)
    const int n4 = (Bsz * Nn) / 4;                 // float4 count
    zero_kernel<<<2048, 256, 0, stream>>>((float4*)depth_out, n4);

    // 2) transform + project + scatter
    dim3 grid(256, Bsz);                           // 256 blocks/batch, 8 waves each
    st_kernel<<<grid, 256, 0, stream>>>(quat, radar, kmat, trans,
                                        depth_out, cloud_out);
}